// Model_49117245997365
// MI455X (gfx1250) — compile-verified
//
#include <hip/hip_runtime.h>
#include <math.h>

typedef __attribute__((ext_vector_type(2))) float v2f;
typedef __attribute__((ext_vector_type(8))) float v8f;

#define P 113
#define D 128
#define H 4
#define DH 32
#define HID 512
#define BATCH 131072
#define NPAIR (P*P)                /* 12769 */
#define NROWS (2*NPAIR)            /* 25538 */
#define NTILES_PAD 1600
#define NROWS_PAD (NTILES_PAD*16)  /* 25600 */

/* workspace layout (float offsets) */
#define OFF_R    0
#define SZ_R     (2*P*D)           /* 28928 */
#define OFF_Q    (OFF_R + SZ_R)
#define OFF_K    (OFF_Q + SZ_R)
#define OFF_VO   (OFF_K + SZ_R)
#define SZ_VO    (2*P*H*D)         /* 115712 */
#define OFF_RM   (OFF_VO + SZ_VO)
#define SZ_RM    (NROWS_PAD*D)
#define OFF_TAB  (OFF_RM + SZ_RM)

// -------------------- Phase 1: per-token precompute --------------------
// For each (token t, position pos): resid row, q/k vectors (flattened h*32+k),
// and per-head v @ W_O projections.
__global__ void __launch_bounds__(128) prep_kernel(
    const float* __restrict__ tok_emb, const float* __restrict__ pos_emb,
    const float* __restrict__ W_Q, const float* __restrict__ W_K,
    const float* __restrict__ W_V, const float* __restrict__ W_O,
    float* __restrict__ ws) {
  int t = blockIdx.x, pos = blockIdx.y, i = threadIdx.x;   // 128 threads
  float* R  = ws + OFF_R;
  float* Q  = ws + OFF_Q;
  float* K  = ws + OFF_K;
  float* VO = ws + OFF_VO;
  __shared__ float rs[D], vs[D];
  float r = tok_emb[t*D + i] + pos_emb[pos*D + i];
  int row = pos*P + t;
  R[row*D + i] = r;
  rs[i] = r;
  __syncthreads();
  int h = i >> 5, k = i & 31;                              // col i = h*32+k
  float q = 0.f, kk = 0.f, v = 0.f;
  #pragma unroll 8
  for (int d = 0; d < D; ++d) {
    float rv = rs[d];
    int wi = h*(D*DH) + d*DH + k;                          // W_[h][d][k]
    q  += rv * W_Q[wi];
    kk += rv * W_K[wi];
    v  += rv * W_V[wi];
  }
  Q[row*D + i] = q;
  K[row*D + i] = kk;
  vs[i] = v;
  __syncthreads();
  #pragma unroll
  for (int h2 = 0; h2 < H; ++h2) {                         // VO[row][h2][d=i]
    float acc = 0.f;
    #pragma unroll 8
    for (int k2 = 0; k2 < DH; ++k2)
      acc += vs[h2*DH + k2] * W_O[(h2*DH + k2)*D + i];     // W_O[h][k][d]
    VO[(row*H + h2)*D + i] = acc;
  }
}

// -------------------- Phase 2: per-pair attention combine --------------------
__global__ void __launch_bounds__(128) pair_kernel(float* __restrict__ ws) {
  int n = blockIdx.x, i = threadIdx.x;                     // 128 threads
  int a = n / P, c = n % P;
  const float* R  = ws + OFF_R;
  const float* Q  = ws + OFF_Q;
  const float* K  = ws + OFF_K;
  const float* VO = ws + OFF_VO;
  float* Rm = ws + OFF_RM;
  __shared__ float q0s[D], k0s[D], q1s[D], k1s[D];
  __shared__ float ssh[2][H][2];
  __shared__ float pat[2][H][2];
  q0s[i] = Q[a*D + i];        k0s[i] = K[a*D + i];
  q1s[i] = Q[(P + c)*D + i];  k1s[i] = K[(P + c)*D + i];
  __syncthreads();
  if (i < 16) {                                            // 16 dot products
    int qp = i >> 3, h = (i >> 1) & 3, sp = i & 1;
    const float* qv = qp ? q1s : q0s;
    const float* kv = sp ? k1s : k0s;
    float s = 0.f;
    #pragma unroll 8
    for (int k = 0; k < DH; ++k) s += qv[h*DH + k] * kv[h*DH + k];
    ssh[qp][h][sp] = s * 0.17677669529663687f;             // 1/sqrt(32)
  }
  __syncthreads();
  if (i < 8) {                                             // 2-way softmax + blend
    int qp = i >> 2, h = i & 3;
    float s0 = ssh[qp][h][0], s1 = ssh[qp][h][1];
    float m = fmaxf(s0, s1);
    float e0 = __expf(s0 - m), e1 = __expf(s1 - m);
    float inv = 1.f / (e0 + e1);
    pat[qp][h][0] = 0.5f + 0.5f * (e0 * inv);
    pat[qp][h][1] = 0.5f + 0.5f * (e1 * inv);
  }
  __syncthreads();
  float o0 = R[a*D + i];
  float o1 = R[(P + c)*D + i];
  #pragma unroll
  for (int h = 0; h < H; ++h) {
    float voa = VO[(a*H + h)*D + i];                       // pos0 token a
    float voc = VO[((P + c)*H + h)*D + i];                 // pos1 token c
    o0 += pat[0][h][0]*voa + pat[0][h][1]*voc;
    o1 += pat[1][h][0]*voa + pat[1][h][1]*voc;
  }
  Rm[(2*n + 0)*D + i] = o0;
  Rm[(2*n + 1)*D + i] = o1;
}

// -------------------- Phase 3: fused MLP + unembed via f32 WMMA --------------------
__device__ __forceinline__ v8f wmma4(v2f a, v2f b, v8f c) {
  return __builtin_amdgcn_wmma_f32_16x16x4_f32(false, a, false, b, (short)0, c,
                                               false, false);
}

#define LS 132                                             /* LDS row stride */

__global__ void __launch_bounds__(128) mlp_kernel(
    const float* __restrict__ W1, const float* __restrict__ b1,
    const float* __restrict__ W2, const float* __restrict__ b2,
    const float* __restrict__ WU, float* __restrict__ ws) {
  __shared__ float lds[4][16*LS];                          // 33.8 KB, 4 waves
  const float* Rm = ws + OFF_RM;
  float* tab = ws + OFF_TAB;
  int wid  = threadIdx.x >> 5;
  int lane = threadIdx.x & 31;
  int half = lane >> 4, lm = lane & 15;
  int row0 = (blockIdx.x*4 + wid) * 16;
  float* L = &lds[wid][0];

  // A-fragments for resid_mid rows, K = 128 (kept resident)
  v2f a1[32];
  const float* Ar = Rm + (size_t)(row0 + lm)*D + 2*half;
  #pragma unroll
  for (int kk = 0; kk < 32; ++kk) {
    v2f t; t.x = Ar[4*kk]; t.y = Ar[4*kk + 1];
    a1[kk] = t;
  }
  v8f out2[8];
  #pragma unroll
  for (int n2 = 0; n2 < 8; ++n2) out2[n2] = {};

  for (int k2 = 0; k2 < 32; ++k2) {                        // 512 hidden / 16
    // ---- GEMM1: h_tile[16x16] = A(16x128) @ W1[:, k2*16..] ----
    v8f hh = {};
    int colb = k2*16 + lm;
    #pragma unroll
    for (int kk = 0; kk < 32; ++kk) {
      int kr = 4*kk + 2*half;
      v2f b; b.x = W1[kr*HID + colb]; b.y = W1[(kr+1)*HID + colb];
      hh = wmma4(a1[kk], b, hh);
    }
    float bias = b1[colb];
    #pragma unroll
    for (int r = 0; r < 8; ++r) {                          // bias + relu -> LDS
      float val = fmaxf(hh[r] + bias, 0.f);
      L[(r + 8*half)*LS + lm] = val;                       // C layout (M,N)
    }
    __builtin_amdgcn_wave_barrier();
    asm volatile("s_wait_dscnt 0" ::: "memory");
    v2f a2[4];                                             // re-read as A frags
    #pragma unroll
    for (int j = 0; j < 4; ++j) {
      a2[j].x = L[lm*LS + 4*j + 2*half];
      a2[j].y = L[lm*LS + 4*j + 1 + 2*half];
    }
    __builtin_amdgcn_wave_barrier();
    // ---- GEMM2: out2[16x128] += h_tile @ W2[k2*16.., :] ----
    #pragma unroll
    for (int j = 0; j < 4; ++j) {
      int kr = k2*16 + 4*j + 2*half;
      #pragma unroll
      for (int n2 = 0; n2 < 8; ++n2) {
        v2f b; b.x = W2[kr*D + n2*16 + lm]; b.y = W2[(kr+1)*D + n2*16 + lm];
        out2[n2] = wmma4(a2[j], b, out2[n2]);
      }
    }
  }

  // ---- skip connection + b2; stage final resid tile (16x128) into LDS ----
  #pragma unroll
  for (int n2 = 0; n2 < 8; ++n2) {
    int col = n2*16 + lm;
    float bb = b2[col];
    #pragma unroll
    for (int r = 0; r < 8; ++r) {
      int row = row0 + r + 8*half;
      float val = out2[n2][r] + bb + Rm[(size_t)row*D + col];
      L[(r + 8*half)*LS + col] = val;
    }
  }
  __builtin_amdgcn_wave_barrier();
  asm volatile("s_wait_dscnt 0" ::: "memory");
  v2f a3[32];
  #pragma unroll
  for (int kk = 0; kk < 32; ++kk) {
    a3[kk].x = L[lm*LS + 4*kk + 2*half];
    a3[kk].y = L[lm*LS + 4*kk + 1 + 2*half];
  }
  // ---- GEMM3: logits[16x113] = resid @ W_U^T  (B frag = W_U rows, contiguous K) ----
  #pragma unroll 1
  for (int n3 = 0; n3 < 8; ++n3) {
    int v = n3*16 + lm;
    int vc = (v < P) ? v : 0;
    const float* Wr = WU + vc*D + 2*half;
    v8f acc = {};
    #pragma unroll
    for (int kk = 0; kk < 32; ++kk) {
      v2f b; b.x = Wr[4*kk]; b.y = Wr[4*kk + 1];
      acc = wmma4(a3[kk], b, acc);
    }
    #pragma unroll
    for (int r = 0; r < 8; ++r) {
      int row = row0 + r + 8*half;
      if (row < NROWS && v < P) tab[(size_t)row*P + v] = acc[r];
    }
  }
}

// -------------------- Phase 4: gather (bandwidth-bound) --------------------
__global__ void __launch_bounds__(256) gather_kernel(
    const int* __restrict__ x, const float* __restrict__ tab,
    float* __restrict__ out) {
  int b = blockIdx.x;
  int t = threadIdx.x;
  int a = x[2*b], c = x[2*b + 1];
  int n = a*P + c;
  if (t < 2*P)                                             // table row pair is contiguous
    out[(size_t)b*(2*P) + t] = tab[(size_t)n*(2*P) + t];
}

extern "C" void kernel_launch(void* const* d_in, const int* in_sizes, int n_in,
                              void* d_out, int out_size, void* d_ws, size_t ws_size,
                              hipStream_t stream) {
  (void)in_sizes; (void)n_in; (void)out_size; (void)ws_size;
  const int*   x   = (const int*)d_in[0];
  const float* tok = (const float*)d_in[1];
  const float* pos = (const float*)d_in[2];
  const float* W_Q = (const float*)d_in[3];
  const float* W_K = (const float*)d_in[4];
  const float* W_V = (const float*)d_in[5];
  const float* W_O = (const float*)d_in[6];
  const float* W1  = (const float*)d_in[7];
  const float* b1  = (const float*)d_in[8];
  const float* W2  = (const float*)d_in[9];
  const float* b2  = (const float*)d_in[10];
  const float* WU  = (const float*)d_in[11];
  float* ws  = (float*)d_ws;
  float* out = (float*)d_out;

  prep_kernel<<<dim3(P, 2), D, 0, stream>>>(tok, pos, W_Q, W_K, W_V, W_O, ws);
  pair_kernel<<<NPAIR, D, 0, stream>>>(ws);
  mlp_kernel<<<NTILES_PAD/4, 128, 0, stream>>>(W1, b1, W2, b2, WU, ws);
  gather_kernel<<<BATCH, 256, 0, stream>>>(x, ws + OFF_TAB, out);
}